// CPPrefill_66709432041911
// MI455X (gfx1250) — compile-verified
//
#include <hip/hip_runtime.h>
#include <math.h>

// Problem constants (from reference): L,D,H,HD,FF,V,NH,S
#define LAYERS   5
#define DMODEL   1024
#define NHEADS   8
#define HEADDIM  128
#define FFDIM    4096
#define VOCAB    2048
#define NLMHEADS 15
#define SEQ      2
#define RMS_EPS  1e-6f

typedef __attribute__((ext_vector_type(2))) float v2f;
typedef __attribute__((ext_vector_type(8))) float v8f;

#if defined(__gfx1250__) && __has_builtin(__builtin_amdgcn_wmma_f32_16x16x4_f32)
#define HAVE_WMMA_F32X4 1
#else
#define HAVE_WMMA_F32X4 0
#endif

// ---------------------------------------------------------------------------
// h = concat(past_hidden, codec_emb[cb0_token]) : (2 x D)
// ---------------------------------------------------------------------------
__global__ void build_h_kernel(const float* __restrict__ past,
                               const float* __restrict__ emb,
                               const int* __restrict__ tok,
                               float* __restrict__ h) {
  const int t = tok[0];
  for (int i = threadIdx.x; i < DMODEL; i += blockDim.x) {
    h[i]          = past[i];
    h[DMODEL + i] = emb[(long)t * DMODEL + i];
  }
}

// ---------------------------------------------------------------------------
// RMSNorm over last dim; one block per row (wave32-aware reduction)
// ---------------------------------------------------------------------------
__global__ void rmsnorm_kernel(const float* __restrict__ x,
                               const float* __restrict__ w,
                               float* __restrict__ y, int D) {
  const float* xr = x + (long)blockIdx.x * D;
  float*       yr = y + (long)blockIdx.x * D;
  __shared__ float sh[8];
  float ss = 0.f;
  for (int i = threadIdx.x; i < D; i += blockDim.x) { float t = xr[i]; ss += t * t; }
  for (int o = 16; o > 0; o >>= 1) ss += __shfl_down(ss, o, 32);  // wave32
  if ((threadIdx.x & 31) == 0) sh[threadIdx.x >> 5] = ss;
  __syncthreads();
  float tot = 0.f;
  const int nw = blockDim.x >> 5;
  for (int i = 0; i < nw; ++i) tot += sh[i];
  const float inv = rsqrtf(tot / (float)D + RMS_EPS);
  for (int i = threadIdx.x; i < D; i += blockDim.x) yr[i] = xr[i] * inv * w[i];
}

// ---------------------------------------------------------------------------
// Y[M x N] (+= addsrc) = X[M x K] @ W[K x N], M in {1,2} real rows padded to
// the 16-row WMMA tile, via V_WMMA_F32_16X16X4_F32 (fp32 weights stay fp32:
// the workload is HBM-bound, conversion saves no bytes).
//
// K,N are compile-time so the k*N terms fold into immediate offsets; X is
// staged into LDS with an explicit zero row so the A fragment is one
// branch-free ds_load_b64 per step (no exec-mask predication); W is streamed
// with non-temporal loads (read-once, keep L2 for activations).
//
// VGPR fragment layouts per CDNA5 ISA 7.12.2 (wave32):
//   A 16x4:  lane l, vgpr j -> A[M=l%16][K = 2*(l/16)+j]
//   B 4x16:  lane l, vgpr j -> B[K = j+2*(l/16)][N = n0+l%16]
//   C 16x16: lane l, vgpr r -> C[M = r+8*(l/16)][N = n0+l%16]
// ---------------------------------------------------------------------------
template <int K, int N>
__global__ __launch_bounds__(128)
void gemm_wmma_f32_kernel(const float* __restrict__ X,
                          const float* __restrict__ W,
                          const float* addsrc,   // optional residual (may alias Y)
                          float* Y,
                          int Mrows, long wStride, long yStride) {
  __shared__ __align__(16) float shx[3 * K];  // rows 0,1 = X; row 2 = zeros
  for (int i = threadIdx.x; i < K; i += 128) {
    shx[i]         = X[i];
    shx[K + i]     = (Mrows > 1) ? X[K + i] : 0.f;
    shx[2 * K + i] = 0.f;
  }
  __syncthreads();

  const int lane  = threadIdx.x & 31;
  const int wav   = threadIdx.x >> 5;
  const int ntile = (blockIdx.x * 4 + wav) << 4;
  if (ntile >= N) return;

  const int kh = lane >> 4;      // half-wave -> K sub-offset
  const int lm = lane & 15;      // M for A, N for B/C
  const int arow = (lm < Mrows) ? lm : 2;                  // zero row for pads
  const float* ax = &shx[arow * K + 2 * kh];               // 8B aligned
  const float* wp = W + (long)blockIdx.y * wStride
                      + (long)(2 * kh) * N + ntile + lm;

  v8f acc = {};
#pragma unroll 8
  for (int k0 = 0; k0 < K; k0 += 4) {
    const v2f a = *(const v2f*)(ax + k0);                  // ds_load_b64
    v2f b;
    b.x = __builtin_nontemporal_load(wp + (long)k0 * N);       // row k0+2kh
    b.y = __builtin_nontemporal_load(wp + (long)k0 * N + N);   // row k0+2kh+1
#if HAVE_WMMA_F32X4
    acc = __builtin_amdgcn_wmma_f32_16x16x4_f32(
        /*neg_a=*/false, a, /*neg_b=*/false, b,
        /*c_mod=*/(short)0, acc, /*reuse_a=*/false, /*reuse_b=*/false);
#else
    acc[0] += a.x * b.x + a.y * b.y;  // compile fallback only
#endif
  }

  // Rows 0..Mrows-1 live in vgprs 0..Mrows-1 of the lanes with kh==0.
  if (kh == 0) {
    float* Yb = Y + (long)blockIdx.y * yStride;
    const int col = ntile + lm;
    if (Mrows > 0) {
      const float r0 = addsrc ? addsrc[col] : 0.f;
      Yb[col] = acc[0] + r0;
    }
    if (Mrows > 1) {
      const float r1 = addsrc ? addsrc[N + col] : 0.f;
      Yb[N + col] = acc[1] + r1;
    }
  }
}

// ---------------------------------------------------------------------------
// Per-head: q/k RMSNorm, RoPE (pos 0 identity, pos 1), causal S=2 softmax,
// o = w @ v, and write post-rope K / raw V into the kv-cache region of d_out.
// One block (128 threads = 4 waves) per head.
// ---------------------------------------------------------------------------
__device__ __forceinline__ float blocksum128(float v, float* red) {
  for (int o = 16; o > 0; o >>= 1) v += __shfl_down(v, o, 32);
  __syncthreads();
  if ((threadIdx.x & 31) == 0) red[threadIdx.x >> 5] = v;
  __syncthreads();
  return red[0] + red[1] + red[2] + red[3];
}

__global__ __launch_bounds__(128)
void attn_kernel(const float* __restrict__ q_in, const float* __restrict__ k_in,
                 const float* __restrict__ v_in,
                 const float* __restrict__ qn, const float* __restrict__ kn,
                 float* __restrict__ o_out,
                 float* __restrict__ kv_k, float* __restrict__ kv_v) {
  const int h = blockIdx.x;
  const int d = threadIdx.x;
  __shared__ float red[4];
  __shared__ float shq[HEADDIM];
  __shared__ float shk[HEADDIM];
  const int base = h * HEADDIM + d;

  float q0 = q_in[base], q1 = q_in[DMODEL + base];
  float k0 = k_in[base], k1 = k_in[DMODEL + base];
  float v0 = v_in[base], v1 = v_in[DMODEL + base];

  const float sq0 = blocksum128(q0 * q0, red);
  const float sq1 = blocksum128(q1 * q1, red);
  const float sk0 = blocksum128(k0 * k0, red);
  const float sk1 = blocksum128(k1 * k1, red);
  q0 *= rsqrtf(sq0 / (float)HEADDIM + RMS_EPS) * qn[d];
  q1 *= rsqrtf(sq1 / (float)HEADDIM + RMS_EPS) * qn[d];
  k0 *= rsqrtf(sk0 / (float)HEADDIM + RMS_EPS) * kn[d];
  k1 *= rsqrtf(sk1 / (float)HEADDIM + RMS_EPS) * kn[d];

  // RoPE at position 1 (position 0 is identity: cos=1, sin=0)
  shq[d] = q1; shk[d] = k1;
  __syncthreads();
  const float ang = powf(10000.f, -(float)(2 * (d & 63)) / (float)HEADDIM); // pos=1
  const float c = cosf(ang), s = sinf(ang);
  const float q1o = (d < 64) ? -shq[d + 64] : shq[d - 64];
  const float k1o = (d < 64) ? -shk[d + 64] : shk[d - 64];
  const float q1r = q1 * c + q1o * s;
  const float k1r = k1 * c + k1o * s;

  // Causal attention: row 0 attends only to k0 -> o0 = v0.
  const float scale = 0.08838834764831845f;  // 1/sqrt(128)
  const float s10 = blocksum128(q1r * k0,  red) * scale;
  const float s11 = blocksum128(q1r * k1r, red) * scale;
  const float mx = fmaxf(s10, s11);
  const float e0 = expf(s10 - mx), e1 = expf(s11 - mx);
  const float inv = 1.f / (e0 + e1);
  o_out[base]          = v0;
  o_out[DMODEL + base] = (e0 * v0 + e1 * v1) * inv;

  // kv_cache layout: [..., H, S, HD]
  kv_k[h * (SEQ * HEADDIM) + d]           = k0;   // pos0 rope is identity
  kv_k[h * (SEQ * HEADDIM) + HEADDIM + d] = k1r;
  kv_v[h * (SEQ * HEADDIM) + d]           = v0;
  kv_v[h * (SEQ * HEADDIM) + HEADDIM + d] = v1;
}

// ---------------------------------------------------------------------------
// y = silu(g) * u
// ---------------------------------------------------------------------------
__global__ void silu_mul_kernel(const float* __restrict__ g,
                                const float* __restrict__ u,
                                float* __restrict__ y, int n) {
  const int i = blockIdx.x * blockDim.x + threadIdx.x;
  if (i < n) { const float x = g[i]; y[i] = (x / (1.f + expf(-x))) * u[i]; }
}

// ---------------------------------------------------------------------------
extern "C" void kernel_launch(void* const* d_in, const int* in_sizes, int n_in,
                              void* d_out, int out_size, void* d_ws, size_t ws_size,
                              hipStream_t stream) {
  (void)in_sizes; (void)n_in; (void)out_size; (void)ws_size;
  const float* past = (const float*)d_in[0];
  const int*   tok  = (const int*)d_in[1];
  const float* emb  = (const float*)d_in[2];
  const float* Wq   = (const float*)d_in[3];
  const float* Wk   = (const float*)d_in[4];
  const float* Wv   = (const float*)d_in[5];
  const float* Wo   = (const float*)d_in[6];
  const float* qnw  = (const float*)d_in[7];
  const float* knw  = (const float*)d_in[8];
  const float* ln1  = (const float*)d_in[9];
  const float* ln2  = (const float*)d_in[10];
  const float* Wg   = (const float*)d_in[11];
  const float* Wu   = (const float*)d_in[12];
  const float* Wd   = (const float*)d_in[13];
  const float* fnw  = (const float*)d_in[14];
  const float* lmh  = (const float*)d_in[15];

  float* ws   = (float*)d_ws;
  float* h    = ws;                    // 2*D
  float* xn   = ws + 2 * DMODEL;       // 2*D
  float* qb   = ws + 4 * DMODEL;       // 2*D
  float* kb   = ws + 6 * DMODEL;       // 2*D
  float* vb   = ws + 8 * DMODEL;       // 2*D
  float* ao   = ws + 10 * DMODEL;      // 2*D
  float* gate = ws + 12 * DMODEL;      // 2*FF
  float* up   = gate + 2 * FFDIM;      // 2*FF
  float* act  = up + 2 * FFDIM;        // 2*FF
  float* last = act + 2 * FFDIM;       // D

  float* logits = (float*)d_out;                 // NH * V
  float* kv     = logits + NLMHEADS * VOCAB;     // 2L * H * S * HD

  build_h_kernel<<<1, 256, 0, stream>>>(past, emb, tok, h);

  const long QKN = (long)NHEADS * HEADDIM;  // 1024
  for (int l = 0; l < LAYERS; ++l) {
    const float* wq = Wq + (long)l * DMODEL * QKN;
    const float* wk = Wk + (long)l * DMODEL * QKN;
    const float* wv = Wv + (long)l * DMODEL * QKN;
    const float* wo = Wo + (long)l * QKN * DMODEL;
    const float* wg = Wg + (long)l * DMODEL * FFDIM;
    const float* wu = Wu + (long)l * DMODEL * FFDIM;
    const float* wd = Wd + (long)l * FFDIM * DMODEL;

    rmsnorm_kernel<<<2, 256, 0, stream>>>(h, ln1 + (long)l * DMODEL, xn, DMODEL);

    // QKV projections: N=1024 -> 64 tiles -> 16 blocks x 4 waves
    gemm_wmma_f32_kernel<DMODEL, DMODEL><<<dim3(16, 1), 128, 0, stream>>>(
        xn, wq, nullptr, qb, SEQ, 0, 0);
    gemm_wmma_f32_kernel<DMODEL, DMODEL><<<dim3(16, 1), 128, 0, stream>>>(
        xn, wk, nullptr, kb, SEQ, 0, 0);
    gemm_wmma_f32_kernel<DMODEL, DMODEL><<<dim3(16, 1), 128, 0, stream>>>(
        xn, wv, nullptr, vb, SEQ, 0, 0);

    attn_kernel<<<NHEADS, HEADDIM, 0, stream>>>(
        qb, kb, vb, qnw + (long)l * HEADDIM, knw + (long)l * HEADDIM, ao,
        kv + (long)(2 * l) * NHEADS * SEQ * HEADDIM,
        kv + (long)(2 * l + 1) * NHEADS * SEQ * HEADDIM);

    // Output projection fused with residual add: h = h + ao @ Wo
    gemm_wmma_f32_kernel<DMODEL, DMODEL><<<dim3(16, 1), 128, 0, stream>>>(
        ao, wo, h, h, SEQ, 0, 0);

    rmsnorm_kernel<<<2, 256, 0, stream>>>(h, ln2 + (long)l * DMODEL, xn, DMODEL);

    // MLP: gate/up (N=4096 -> 256 tiles -> 64 blocks), silu*up, down + residual
    gemm_wmma_f32_kernel<DMODEL, FFDIM><<<dim3(64, 1), 128, 0, stream>>>(
        xn, wg, nullptr, gate, SEQ, 0, 0);
    gemm_wmma_f32_kernel<DMODEL, FFDIM><<<dim3(64, 1), 128, 0, stream>>>(
        xn, wu, nullptr, up, SEQ, 0, 0);
    silu_mul_kernel<<<(2 * FFDIM + 255) / 256, 256, 0, stream>>>(gate, up, act, 2 * FFDIM);
    gemm_wmma_f32_kernel<FFDIM, DMODEL><<<dim3(16, 1), 128, 0, stream>>>(
        act, wd, h, h, SEQ, 0, 0);
  }

  // Final norm on the last token, then 15 batched LM heads (gridDim.y).
  rmsnorm_kernel<<<1, 256, 0, stream>>>(h + DMODEL, fnw, last, DMODEL);
  gemm_wmma_f32_kernel<DMODEL, VOCAB><<<dim3(32, NLMHEADS), 128, 0, stream>>>(
      last, lmh, nullptr, logits, 1, (long)DMODEL * VOCAB, (long)VOCAB);
}